// GATEncoder_5652176962236
// MI455X (gfx1250) — compile-verified
//
#include <hip/hip_runtime.h>

// Problem constants (match reference)
#define Nn  50000
#define Ee  800000
#define Hh  4
#define Dd  128
#define HDc 512
#define Ll  4
#define NEG_ATTN 0.2f
#define NEG_ACT  0.01f
#define LN_EPS   1e-5f

typedef float v2f __attribute__((ext_vector_type(2)));
typedef float v8f __attribute__((ext_vector_type(8)));

__device__ __forceinline__ float lrelu(float x, float s) { return x > 0.f ? x : s * x; }

// float atomic-max via sign-aware integer atomics (native VMEM atomics, no CAS loop)
__device__ __forceinline__ void atomicMaxF(float* a, float v) {
  if (v >= 0.f) atomicMax((int*)a, __float_as_int(v));
  else          atomicMin((unsigned int*)a, __float_as_uint(v));
}

// ---------------------------------------------------------------------------
// init per-node softmax state
__global__ void k_init(float* __restrict__ m, float* __restrict__ z) {
  int i = blockIdx.x * 256 + threadIdx.x;
  if (i < Nn * Hh) { m[i] = -__builtin_inff(); z[i] = 0.f; }
}

// ---------------------------------------------------------------------------
// out[N,512] = X[N,128] @ W[128,512] (+ bias[512] if non-null)
// block = 256 thr (8 waves), one 16-row M tile per block; wave w covers cols
// [w*64, w*64+64) as 4 WMMA n-subtiles. A tile staged in LDS, stride 132
// (132%64==4 -> fragment rows land on distinct banks).
__global__ void k_gemm_k128(const float* __restrict__ X, const float* __restrict__ W,
                            const float* __restrict__ bias, float* __restrict__ out) {
  __shared__ float As[16 * 132];
  const int m0 = blockIdx.x * 16;
  const int t  = threadIdx.x;
  { // stage A: 2048 floats, 8 per thread
    int flat = t * 8;
    int r = flat >> 7, c = flat & 127;
    float4 a0 = *(const float4*)(X + (size_t)(m0 + r) * Dd + c);
    float4 a1 = *(const float4*)(X + (size_t)(m0 + r) * Dd + c + 4);
    *(float4*)&As[r * 132 + c]     = a0;
    *(float4*)&As[r * 132 + c + 4] = a1;
  }
  __syncthreads();

  const int wave = t >> 5;
  const int lane = t & 31;
  const int half = lane >> 4;   // 0: K lo pair, 1: K hi pair
  const int mr   = lane & 15;
  const int n0   = wave * 64;

  v8f acc[4] = {};
  for (int k = 0; k < Dd; k += 4) {
    const int ka = k + 2 * half;
    v2f a = *(const v2f*)&As[mr * 132 + ka];
#pragma unroll
    for (int s = 0; s < 4; ++s) {
      const int col = n0 + s * 16 + mr;
      v2f b;
      b.x = W[(size_t)ka * HDc + col];
      b.y = W[(size_t)(ka + 1) * HDc + col];
      acc[s] = __builtin_amdgcn_wmma_f32_16x16x4_f32(false, a, false, b,
                                                     (short)0, acc[s], false, false);
    }
  }
#pragma unroll
  for (int s = 0; s < 4; ++s) {
    const int col = n0 + s * 16 + mr;
    const float bi = bias ? bias[col] : 0.f;
#pragma unroll
    for (int v = 0; v < 8; ++v) {
      const int row = m0 + v + 8 * half;
      out[(size_t)row * HDc + col] = acc[s][v] + bi;
    }
  }
}

// ---------------------------------------------------------------------------
// el/er[N,H]: one wave per node; lane covers 16 contiguous cols (one head = 8 lanes)
__global__ void k_attn(const float* __restrict__ feat, const float* __restrict__ al,
                       const float* __restrict__ ar, float* __restrict__ el,
                       float* __restrict__ er) {
  const int wave = (blockIdx.x * blockDim.x + threadIdx.x) >> 5;
  const int lane = threadIdx.x & 31;
  const int c0 = lane * 16;
  float sl = 0.f, sr = 0.f;
#pragma unroll
  for (int j = 0; j < 16; j += 4) {
    float4 f = *(const float4*)(feat + (size_t)wave * HDc + c0 + j);
    float4 a = *(const float4*)(al + c0 + j);
    float4 b = *(const float4*)(ar + c0 + j);
    sl += f.x * a.x + f.y * a.y + f.z * a.z + f.w * a.w;
    sr += f.x * b.x + f.y * b.y + f.z * b.z + f.w * b.w;
  }
#pragma unroll
  for (int msk = 1; msk <= 4; msk <<= 1) {
    sl += __shfl_xor(sl, msk, 32);
    sr += __shfl_xor(sr, msk, 32);
  }
  if ((lane & 7) == 0) {
    const int h = lane >> 3;
    el[(size_t)wave * Hh + h] = sl;
    er[(size_t)wave * Hh + h] = sr;
  }
}

// ---------------------------------------------------------------------------
// e = leaky(el[src]+er[dst]); segment max via atomics
__global__ void k_edge_logits(const int* __restrict__ src, const int* __restrict__ dst,
                              const float* __restrict__ el, const float* __restrict__ er,
                              float* __restrict__ ebuf, float* __restrict__ m) {
  const int i = blockIdx.x * 256 + threadIdx.x;
  const int s = src[i], d = dst[i];
  float4 a = *(const float4*)(el + (size_t)s * Hh);
  float4 b = *(const float4*)(er + (size_t)d * Hh);
  float4 e;
  e.x = lrelu(a.x + b.x, NEG_ATTN);
  e.y = lrelu(a.y + b.y, NEG_ATTN);
  e.z = lrelu(a.z + b.z, NEG_ATTN);
  e.w = lrelu(a.w + b.w, NEG_ATTN);
  *(float4*)(ebuf + (size_t)i * Hh) = e;
  atomicMaxF(m + (size_t)d * Hh + 0, e.x);
  atomicMaxF(m + (size_t)d * Hh + 1, e.y);
  atomicMaxF(m + (size_t)d * Hh + 2, e.z);
  atomicMaxF(m + (size_t)d * Hh + 3, e.w);
}

// ex = exp(e - m[dst]); segment sum via atomics
__global__ void k_edge_exp(const int* __restrict__ dst, float* __restrict__ ebuf,
                           const float* __restrict__ m, float* __restrict__ z) {
  const int i = blockIdx.x * 256 + threadIdx.x;
  const int d = dst[i];
  float4 e  = *(const float4*)(ebuf + (size_t)i * Hh);
  float4 mm = *(const float4*)(m + (size_t)d * Hh);
  float4 ex;
  ex.x = __expf(e.x - mm.x);
  ex.y = __expf(e.y - mm.y);
  ex.z = __expf(e.z - mm.z);
  ex.w = __expf(e.w - mm.w);
  *(float4*)(ebuf + (size_t)i * Hh) = ex;
  atomicAdd(z + (size_t)d * Hh + 0, ex.x);
  atomicAdd(z + (size_t)d * Hh + 1, ex.y);
  atomicAdd(z + (size_t)d * Hh + 2, ex.z);
  atomicAdd(z + (size_t)d * Hh + 3, ex.w);
}

// message passing: one wave per edge; lane covers 16 contiguous cols
__global__ void k_edge_msg(const int* __restrict__ src, const int* __restrict__ dst,
                           const float* __restrict__ ebuf, const float* __restrict__ z,
                           const float* __restrict__ feat, float* __restrict__ rst) {
  const int i = (blockIdx.x * blockDim.x + threadIdx.x) >> 5;
  const int lane = threadIdx.x & 31;
  const int s = src[i], d = dst[i];
  const int h = lane >> 3;
  const float ex = ebuf[(size_t)i * Hh + h];
  const float zz = z[(size_t)d * Hh + h];
  const float alpha = ex / fmaxf(zz, 1e-20f);
  const int c0 = lane * 16;
  const float* fs = feat + (size_t)s * HDc + c0;
  float* rd = rst + (size_t)d * HDc + c0;
#pragma unroll
  for (int j = 0; j < 16; j += 4) {
    float4 f = *(const float4*)(fs + j);
    atomicAdd(rd + j + 0, f.x * alpha);
    atomicAdd(rd + j + 1, f.y * alpha);
    atomicAdd(rd + j + 2, f.z * alpha);
    atomicAdd(rd + j + 3, f.w * alpha);
  }
}

// ---------------------------------------------------------------------------
// xout[N,128] = LN( leaky(rst,0.01)[N,512] @ Wn[512,128] + bn ) * g + b
// 8 waves: wave w -> cols [w*16, w*16+16), K=512. Fused bias + layernorm.
__global__ void k_gemm_nrm_ln(const float* __restrict__ Rst, const float* __restrict__ Wn,
                              const float* __restrict__ bn, const float* __restrict__ g,
                              const float* __restrict__ bln, float* __restrict__ xout) {
  __shared__ float As[16 * 516];
  __shared__ float Y[16 * 132];
  const int m0 = blockIdx.x * 16;
  const int t  = threadIdx.x;
  // stage leaky(rst): 8192 floats, 8x float4 per thread
#pragma unroll
  for (int it = 0; it < 8; ++it) {
    const int idx4 = t + it * 256;
    const int flat = idx4 * 4;
    const int r = flat >> 9, c = flat & 511;
    float4 v = *(const float4*)(Rst + (size_t)(m0 + r) * HDc + c);
    v.x = lrelu(v.x, NEG_ACT); v.y = lrelu(v.y, NEG_ACT);
    v.z = lrelu(v.z, NEG_ACT); v.w = lrelu(v.w, NEG_ACT);
    *(float4*)&As[r * 516 + c] = v;
  }
  __syncthreads();

  const int wave = t >> 5;
  const int lane = t & 31;
  const int half = lane >> 4;
  const int mr   = lane & 15;
  const int n0   = wave * 16;

  v8f acc = {};
  for (int k = 0; k < HDc; k += 4) {
    const int ka = k + 2 * half;
    v2f a = *(const v2f*)&As[mr * 516 + ka];
    v2f b;
    b.x = Wn[(size_t)ka * Dd + n0 + mr];
    b.y = Wn[(size_t)(ka + 1) * Dd + n0 + mr];
    acc = __builtin_amdgcn_wmma_f32_16x16x4_f32(false, a, false, b,
                                                (short)0, acc, false, false);
  }
  // deposit y (+bias) into LDS for the cross-wave layernorm
  {
    const float bi = bn[n0 + mr];
#pragma unroll
    for (int v = 0; v < 8; ++v)
      Y[(v + 8 * half) * 132 + n0 + mr] = acc[v] + bi;
  }
  __syncthreads();

  // waves 0..3 each normalize 4 rows; 32 lanes x 4 cols per row
  if (wave < 4) {
    const int c0 = lane * 4;
    float4 gg = *(const float4*)(g + c0);
    float4 bb = *(const float4*)(bln + c0);
#pragma unroll
    for (int it = 0; it < 4; ++it) {
      const int row = wave * 4 + it;
      float4 y = *(const float4*)&Y[row * 132 + c0];
      float s = y.x + y.y + y.z + y.w;
#pragma unroll
      for (int msk = 1; msk <= 16; msk <<= 1) s += __shfl_xor(s, msk, 32);
      const float mu = s * (1.f / Dd);
      float4 dyv; dyv.x = y.x - mu; dyv.y = y.y - mu; dyv.z = y.z - mu; dyv.w = y.w - mu;
      float v2 = dyv.x * dyv.x + dyv.y * dyv.y + dyv.z * dyv.z + dyv.w * dyv.w;
#pragma unroll
      for (int msk = 1; msk <= 16; msk <<= 1) v2 += __shfl_xor(v2, msk, 32);
      const float rs = rsqrtf(v2 * (1.f / Dd) + LN_EPS);
      float4 o;
      o.x = dyv.x * rs * gg.x + bb.x;
      o.y = dyv.y * rs * gg.y + bb.y;
      o.z = dyv.z * rs * gg.z + bb.z;
      o.w = dyv.w * rs * gg.w + bb.w;
      *(float4*)(xout + (size_t)(m0 + row) * Dd + c0) = o;
    }
  }
}

// ---------------------------------------------------------------------------
extern "C" void kernel_launch(void* const* d_in, const int* in_sizes, int n_in,
                              void* d_out, int out_size, void* d_ws, size_t ws_size,
                              hipStream_t stream) {
  const float* features = (const float*)d_in[0];
  const int*   src      = (const int*)d_in[1];
  const int*   dst      = (const int*)d_in[2];
  const float* W_fc     = (const float*)d_in[3];
  const float* attn_l   = (const float*)d_in[4];
  const float* attn_r   = (const float*)d_in[5];
  const float* W_res    = (const float*)d_in[6];
  const float* b_gat    = (const float*)d_in[7];
  const float* W_nrm    = (const float*)d_in[8];
  const float* b_nrm    = (const float*)d_in[9];
  const float* ln_g     = (const float*)d_in[10];
  const float* ln_b     = (const float*)d_in[11];
  (void)in_sizes; (void)n_in; (void)out_size; (void)ws_size;

  float* ws   = (float*)d_ws;
  float* feat = ws;                               // N*HD
  float* rst  = feat + (size_t)Nn * HDc;          // N*HD
  float* el   = rst + (size_t)Nn * HDc;           // N*H
  float* er   = el + (size_t)Nn * Hh;             // N*H
  float* mmax = er + (size_t)Nn * Hh;             // N*H
  float* zsum = mmax + (size_t)Nn * Hh;           // N*H
  float* ebuf = zsum + (size_t)Nn * Hh;           // E*H
  float* xa   = ebuf + (size_t)Ee * Hh;           // N*D
  float* xb   = xa + (size_t)Nn * Dd;             // N*D

  const int mblocks = Nn / 16;        // 3125
  const int nwblk   = Nn / 8;         // 6250 (one wave per node)
  const int eblocks = Ee / 256;       // 3125
  const int ewblk   = Ee / 8;         // 100000 (one wave per edge)
  const int iblocks = (Nn * Hh + 255) / 256;

  const float* X = features;
  for (int l = 0; l < Ll; ++l) {
    const float* Wf = W_fc   + (size_t)l * Dd * HDc;
    const float* al = attn_l + (size_t)l * Hh * Dd;
    const float* ar = attn_r + (size_t)l * Hh * Dd;
    const float* Wr = W_res  + (size_t)l * Dd * HDc;
    const float* bg = b_gat  + (size_t)l * HDc;
    const float* Wn = W_nrm  + (size_t)l * HDc * Dd;
    const float* bn = b_nrm  + (size_t)l * Dd;
    const float* g  = ln_g   + (size_t)l * Dd;
    const float* bl = ln_b   + (size_t)l * Dd;
    float* xout = (l == Ll - 1) ? (float*)d_out : ((l & 1) ? xb : xa);

    k_init<<<iblocks, 256, 0, stream>>>(mmax, zsum);
    k_gemm_k128<<<mblocks, 256, 0, stream>>>(X, Wf, nullptr, feat);
    k_attn<<<nwblk, 256, 0, stream>>>(feat, al, ar, el, er);
    k_gemm_k128<<<mblocks, 256, 0, stream>>>(X, Wr, bg, rst);
    k_edge_logits<<<eblocks, 256, 0, stream>>>(src, dst, el, er, ebuf, mmax);
    k_edge_exp<<<eblocks, 256, 0, stream>>>(dst, ebuf, mmax, zsum);
    k_edge_msg<<<ewblk, 256, 0, stream>>>(src, dst, ebuf, zsum, feat, rst);
    k_gemm_nrm_ln<<<mblocks, 256, 0, stream>>>(rst, Wn, bn, g, bl, xout);
    X = xout;
  }
}